// SupervisedGraphSage_8735963480189
// MI455X (gfx1250) — compile-verified
//
#include <hip/hip_runtime.h>

typedef __attribute__((ext_vector_type(2))) float v2f;
typedef __attribute__((ext_vector_type(8))) float v8f;

#define EMBED_DIM 64

// Stage 0: zero the clause output (harness poisons d_out; OR-scatter needs zeros).
__global__ void zero_out_kernel(float* __restrict__ out, int n) {
    int i = blockIdx.x * blockDim.x + threadIdx.x;
    if (i < n) out[i] = 0.0f;
}

// Stage 1+2+3 fused: WMMA matvec (scores = embeds @ weight), threshold,
// OR-scatter into clauses. One wave32 handles one 16-row tile.
//
// V_WMMA_F32_16X16X4_F32 layouts (ISA 7.12.2, 32-bit data):
//   A (16x4): VGPR0 lanes0-15 K=0 / lanes16-31 K=2 ; VGPR1 K=1 / K=3, M = lane%16
//   B (4x16): VGPR0 lanes0-15 K=0 / lanes16-31 K=2 ; VGPR1 K=1 / K=3, N = lane%16
//   D (16x16): VGPR j: lanes0-15 -> M=j, lanes16-31 -> M=8+j, N = lane%16
// We broadcast weight across all 16 B columns, so every D column holds the
// same 16 scores; accumulate K=64 in 16 WMMA steps.
__global__ __launch_bounds__(256) void edge_clause_wmma_kernel(
    const float* __restrict__ embeds,
    const float* __restrict__ weight,
    const float* __restrict__ edge_feature,
    const int*   __restrict__ edge_clause,
    float*       __restrict__ out,
    int n_tiles)
{
    const int lane = threadIdx.x & 31;
    const int wave = (int)((blockIdx.x * blockDim.x + threadIdx.x) >> 5);
    if (wave >= n_tiles) return;           // wave-uniform: EXEC stays all-1s

    const int  hi = lane >> 4;             // 0: lanes 0-15, 1: lanes 16-31
    const int  m  = lane & 15;             // M row within tile
    const long row_base = (long)wave * 16;

    // Per-lane base of this row's embeds, shifted to this lane's K pair.
    const float* __restrict__ arow = embeds + (row_base + m) * EMBED_DIM + 2 * hi;
    const float* __restrict__ wv   = weight + 2 * hi;

    v8f c = {};
    #pragma unroll
    for (int k0 = 0; k0 < EMBED_DIM; k0 += 4) {
        v2f a, b;
        a.x = arow[k0];          // K = k0 + 2*hi      (contiguous 8B -> b64 load)
        a.y = arow[k0 + 1];      // K = k0 + 2*hi + 1
        b.x = wv[k0];            // same K striping, broadcast over N
        b.y = wv[k0 + 1];
        // 8 args: (neg_a, A, neg_b, B, c_mod, C, reuse_a, reuse_b)
        c = __builtin_amdgcn_wmma_f32_16x16x4_f32(
                false, a, false, b, (short)0, c, false, false);
    }

    // Lanes 0 and 16 own 8 rows each (every N column of D is identical).
    if (m == 0) {
        const long r0 = row_base + 8 * hi;
        #pragma unroll
        for (int j = 0; j < 8; ++j) {
            const long r = r0 + j;
            // edge_value = score + (1 - sign)/2 ; predicate: > 0.5
            const float v = c[j] + (1.0f - edge_feature[r]) * 0.5f;
            if (v > 0.5f) out[edge_clause[r]] = 1.0f;   // benign same-value race
        }
    }
}

// Scalar tail for n_edges not a multiple of 16 (empty for the reference sizes).
__global__ void edge_clause_tail_kernel(
    const float* __restrict__ embeds,
    const float* __restrict__ weight,
    const float* __restrict__ edge_feature,
    const int*   __restrict__ edge_clause,
    float*       __restrict__ out,
    long start, long n_edges)
{
    const long r = start + (long)blockIdx.x * blockDim.x + threadIdx.x;
    if (r >= n_edges) return;
    float s = 0.0f;
    #pragma unroll
    for (int k = 0; k < EMBED_DIM; ++k) s += embeds[r * EMBED_DIM + k] * weight[k];
    const float v = s + (1.0f - edge_feature[r]) * 0.5f;
    if (v > 0.5f) out[edge_clause[r]] = 1.0f;
}

extern "C" void kernel_launch(void* const* d_in, const int* in_sizes, int n_in,
                              void* d_out, int out_size, void* d_ws, size_t ws_size,
                              hipStream_t stream) {
    const float* embeds       = (const float*)d_in[0];
    const float* weight       = (const float*)d_in[1];
    const float* edge_feature = (const float*)d_in[2];
    const int*   edge_clause  = (const int*)d_in[3];
    float*       out          = (float*)d_out;

    const long n_edges   = (long)in_sizes[2];   // edge_feature element count
    const int  n_clauses = out_size;            // == n_clauses scalar

    // 1) zero clause outputs
    {
        const int threads = 256;
        const int blocks  = (n_clauses + threads - 1) / threads;
        zero_out_kernel<<<blocks, threads, 0, stream>>>(out, n_clauses);
    }

    // 2) WMMA matvec + threshold + OR-scatter (16 rows per wave, 8 waves/block)
    const long n_tiles = n_edges / 16;
    if (n_tiles > 0) {
        const int threads = 256;                         // 8 wave32 per block
        const long waves_per_block = threads / 32;
        const long blocks = (n_tiles + waves_per_block - 1) / waves_per_block;
        edge_clause_wmma_kernel<<<(int)blocks, threads, 0, stream>>>(
            embeds, weight, edge_feature, edge_clause, out, (int)n_tiles);
    }

    // 3) scalar tail for leftover rows
    const long tail_start = n_tiles * 16;
    if (tail_start < n_edges) {
        const long n_tail = n_edges - tail_start;
        const int threads = 256;
        const int blocks  = (int)((n_tail + threads - 1) / threads);
        edge_clause_tail_kernel<<<blocks, threads, 0, stream>>>(
            embeds, weight, edge_feature, edge_clause, out, tail_start, n_edges);
    }
}